// VTLN_71021579206721
// MI455X (gfx1250) — compile-verified
//
#include <hip/hip_runtime.h>

typedef __attribute__((ext_vector_type(2))) float v2f;
typedef __attribute__((ext_vector_type(8))) float v8f;

#define M_CEP   40
#define T_LEN   2000
#define D_POW   81                    // 2*M+1 alpha powers
#define D_PAD   84                    // padded to multiple of 4
#define N_PAD   48                    // padded rows (3 tiles of 16)
#define K_TOT   (M_CEP * D_POW)       // 3240 (raw a3d row stride)
#define K_PAD   (M_CEP * D_PAD)       // 3360 (packed row stride)
#define NW      4                     // waves per block (wave32)

// ---------------------------------------------------------------------------
// Pre-pack a3d [40][40][81] -> Apk [48][40*84], zero-padded (runs per launch).
// ---------------------------------------------------------------------------
__global__ __launch_bounds__(256) void vtln_prepack_kernel(
    const float* __restrict__ a3d, float* __restrict__ apk)
{
    int idx = blockIdx.x * 256 + threadIdx.x;
    if (idx >= N_PAD * K_PAD) return;
    int n = idx / K_PAD;
    int r = idx - n * K_PAD;
    int m = r / D_PAD;
    int d = r - m * D_PAD;
    float v = 0.0f;
    if (n < M_CEP && d < D_POW)
        v = a3d[((size_t)n * M_CEP + m) * D_POW + d];
    apk[idx] = v;
}

// ---------------------------------------------------------------------------
// Fast path: out = Apk[48 x 3360] @ P[3360 x 16] per (b, t-tile).
// P[k,j] = mcep[m,j] * alpha_j^d synthesized on the fly, k = m*84 + d.
// Inner 21 k-steps fully unrolled -> immediate-offset loads, no index VALU.
// ---------------------------------------------------------------------------
__global__ __launch_bounds__(128) void vtln_wmma_packed_kernel(
    const float* __restrict__ mcep,    // [B, 40, 2000]
    const float* __restrict__ alpha,   // [B, 2000]
    const float* __restrict__ apk,     // [48, 3360] packed A
    float* __restrict__ out)           // [B, 40, 2000]
{
    __shared__ float alpha_s[D_PAD * 16];          // [d][j], rows 81..83 = 0
    __shared__ float mcep_s[M_CEP * 16];           // [m][j]
    __shared__ float red[NW * 3 * 32 * 8];         // cross-wave partials

    const int tid  = threadIdx.x;
    const int wave = tid >> 5;
    const int lane = tid & 31;
    const int j    = lane & 15;
    const int half = lane >> 4;
    const int b    = blockIdx.y;
    const int t0   = blockIdx.x * 16;

    if (tid < 16) {
        float a = alpha[(size_t)b * T_LEN + t0 + tid];
        float p = 1.0f;
        for (int d = 0; d < D_POW; ++d) { alpha_s[d * 16 + tid] = p; p *= a; }
        alpha_s[81 * 16 + tid] = 0.0f;
        alpha_s[82 * 16 + tid] = 0.0f;
        alpha_s[83 * 16 + tid] = 0.0f;
    }
    for (int e = tid; e < M_CEP * 16; e += 128) {
        int m = e >> 4, jj = e & 15;
        mcep_s[e] = mcep[((size_t)b * M_CEP + m) * T_LEN + t0 + jj];
    }
    __syncthreads();

    v8f c0 = {}; v8f c1 = {}; v8f c2 = {};
    const float* abase = apk + (size_t)j * K_PAD + half * 2;  // A row = lane&15
    const float* al    = alpha_s + half * 32 + j;             // alpha^d column

    for (int mi = 0; mi < M_CEP / NW; ++mi) {
        const int   m    = wave + mi * NW;
        const float mval = mcep_s[m * 16 + j];                // hoisted: 1 per 63 WMMAs
        const float* ap  = abase + m * D_PAD;
        #pragma unroll
        for (int i = 0; i < D_PAD / 4; ++i) {                 // 21 steps, all imm offsets
            v2f bf;
            bf.x = mval * al[(i * 4 + 0) * 16];
            bf.y = mval * al[(i * 4 + 1) * 16];
            v2f a0 = *(const v2f*)(ap + i * 4);
            v2f a1 = *(const v2f*)(ap + 16 * K_PAD + i * 4);
            v2f a2 = *(const v2f*)(ap + 32 * K_PAD + i * 4);
            c0 = __builtin_amdgcn_wmma_f32_16x16x4_f32(false, a0, false, bf,
                                                       (short)0, c0, false, false);
            c1 = __builtin_amdgcn_wmma_f32_16x16x4_f32(false, a1, false, bf,
                                                       (short)0, c1, false, false);
            c2 = __builtin_amdgcn_wmma_f32_16x16x4_f32(false, a2, false, bf,
                                                       (short)0, c2, false, false);
        }
    }

    {   // cross-wave reduction of K-split partials
        float* d0p = &red[((wave * 3 + 0) * 32 + lane) * 8];
        float* d1p = &red[((wave * 3 + 1) * 32 + lane) * 8];
        float* d2p = &red[((wave * 3 + 2) * 32 + lane) * 8];
        #pragma unroll
        for (int r = 0; r < 8; ++r) { d0p[r] = c0[r]; d1p[r] = c1[r]; d2p[r] = c2[r]; }
    }
    __syncthreads();

    if (wave < 3) {
        const int mt = wave;
        #pragma unroll
        for (int r = 0; r < 8; ++r) {
            float sum = 0.0f;
            #pragma unroll
            for (int w = 0; w < NW; ++w)
                sum += red[((w * 3 + mt) * 32 + lane) * 8 + r];
            int n = mt * 16 + half * 8 + r;                   // C row = half*8 + r
            if (n < M_CEP)
                out[((size_t)b * M_CEP + n) * T_LEN + t0 + j] = sum;
        }
    }
}

// ---------------------------------------------------------------------------
// Fallback (no workspace): reads raw a3d, per-step k = m*81+d decode.
// ---------------------------------------------------------------------------
__global__ __launch_bounds__(128) void vtln_wmma_fallback_kernel(
    const float* __restrict__ mcep, const float* __restrict__ alpha,
    const float* __restrict__ a3d, float* __restrict__ out)
{
    __shared__ float alpha_s[D_POW * 16];
    __shared__ float mcep_s[M_CEP * 16];
    __shared__ float red[NW * 3 * 32 * 8];

    const int tid  = threadIdx.x;
    const int wave = tid >> 5;
    const int lane = tid & 31;
    const int j    = lane & 15;
    const int half = lane >> 4;
    const int b    = blockIdx.y;
    const int t0   = blockIdx.x * 16;

    if (tid < 16) {
        float a = alpha[(size_t)b * T_LEN + t0 + tid];
        float p = 1.0f;
        for (int d = 0; d < D_POW; ++d) { alpha_s[d * 16 + tid] = p; p *= a; }
    }
    for (int e = tid; e < M_CEP * 16; e += 128) {
        int m = e >> 4, jj = e & 15;
        mcep_s[e] = mcep[((size_t)b * M_CEP + m) * T_LEN + t0 + jj];
    }
    __syncthreads();

    v8f c0 = {}; v8f c1 = {}; v8f c2 = {};
    const float* arow = a3d + (size_t)j * K_TOT;
    const bool   has2 = (32 + j) < M_CEP;

    for (int s = wave; s < K_TOT / 4; s += NW) {
        const int k0 = s * 4 + half * 2;
        unsigned uk = (unsigned)k0;
        unsigned m0 = uk / D_POW, d0 = uk - m0 * D_POW;
        unsigned m1 = m0, d1 = d0 + 1;
        if (d1 == D_POW) { d1 = 0; m1 = m0 + 1; }
        v2f bf;
        bf.x = mcep_s[m0 * 16 + j] * alpha_s[d0 * 16 + j];
        bf.y = mcep_s[m1 * 16 + j] * alpha_s[d1 * 16 + j];
        v2f a0 = *(const v2f*)(arow + k0);
        v2f a1 = *(const v2f*)(arow + 16 * K_TOT + k0);
        v2f a2 = {};
        if (has2) a2 = *(const v2f*)(arow + 32 * K_TOT + k0);
        c0 = __builtin_amdgcn_wmma_f32_16x16x4_f32(false, a0, false, bf, (short)0, c0, false, false);
        c1 = __builtin_amdgcn_wmma_f32_16x16x4_f32(false, a1, false, bf, (short)0, c1, false, false);
        c2 = __builtin_amdgcn_wmma_f32_16x16x4_f32(false, a2, false, bf, (short)0, c2, false, false);
    }

    {
        float* d0p = &red[((wave * 3 + 0) * 32 + lane) * 8];
        float* d1p = &red[((wave * 3 + 1) * 32 + lane) * 8];
        float* d2p = &red[((wave * 3 + 2) * 32 + lane) * 8];
        #pragma unroll
        for (int r = 0; r < 8; ++r) { d0p[r] = c0[r]; d1p[r] = c1[r]; d2p[r] = c2[r]; }
    }
    __syncthreads();

    if (wave < 3) {
        const int mt = wave;
        #pragma unroll
        for (int r = 0; r < 8; ++r) {
            float sum = 0.0f;
            #pragma unroll
            for (int w = 0; w < NW; ++w)
                sum += red[((w * 3 + mt) * 32 + lane) * 8 + r];
            int n = mt * 16 + half * 8 + r;
            if (n < M_CEP)
                out[((size_t)b * M_CEP + n) * T_LEN + t0 + j] = sum;
        }
    }
}

extern "C" void kernel_launch(void* const* d_in, const int* in_sizes, int n_in,
                              void* d_out, int out_size, void* d_ws, size_t ws_size,
                              hipStream_t stream) {
    const float* mcep  = (const float*)d_in[0];   // [B, 40, 2000]
    const float* alpha = (const float*)d_in[1];   // [B, 2000]
    const float* a3d   = (const float*)d_in[2];   // [40, 40, 81]
    float* out = (float*)d_out;

    const int B = in_sizes[1] / T_LEN;            // 16
    dim3 grid(T_LEN / 16, B);                     // 125 x 16 blocks

    const size_t pack_bytes = (size_t)N_PAD * K_PAD * sizeof(float);  // 645 KB
    if (ws_size >= pack_bytes) {
        float* apk = (float*)d_ws;
        const int total  = N_PAD * K_PAD;
        vtln_prepack_kernel<<<(total + 255) / 256, 256, 0, stream>>>(a3d, apk);
        vtln_wmma_packed_kernel<<<grid, 128, 0, stream>>>(mcep, alpha, apk, out);
    } else {
        vtln_wmma_fallback_kernel<<<grid, 128, 0, stream>>>(mcep, alpha, a3d, out);
    }
}